// PIXLayer_32074815767155
// MI455X (gfx1250) — compile-verified
//
#include <hip/hip_runtime.h>

// CDNA5 (gfx1250) wave32 WMMA types
typedef __attribute__((ext_vector_type(2))) float v2f;
typedef __attribute__((ext_vector_type(8))) float v8f;

#define N_PROP 64
#define WAVES_PER_BLOCK 8
#define BLOCK_THREADS (WAVES_PER_BLOCK * 32)

// out[(p*3+x), g] = sum_z px[i_p, x, z]*wi[z,g] + px[j_p, x, z]*wj[z,g]
// M = nPairs*3 rows, N = 64, K = 64. One wave computes a 16x64 tile with
// V_WMMA_F32_16X16X4_F32 (exact fp32, matches reference numerics).
//
// Memory strategy (MI455X):
//  - px (38 MB) + weights live in L2; gathers are L2 traffic, not HBM.
//  - output (614 MB) is the only large HBM stream -> non-temporal stores so
//    it does not thrash px out of the 192 MB L2.
//  - all 32 A-fragment gathers per tile are issued as one batch (clauses) so
//    the tile pays L2 latency once, then runs 128 back-to-back WMMAs.
__global__ __launch_bounds__(BLOCK_THREADS) void pix_wmma_kernel(
    const int*   __restrict__ ind2,   // [nPairs, 2]
    const float* __restrict__ px,     // [nAtoms, 3, 64]
    const float* __restrict__ wi,     // [64, 64]
    const float* __restrict__ wj,     // [64, 64]
    float*       __restrict__ out,    // [nPairs*3, 64]
    int nPairs)
{
    // Stage both weight matrices in LDS (2 * 16 KB of the 320 KB WGP LDS).
    __shared__ float sW[2][N_PROP * N_PROP];
    {
        const float4* a  = (const float4*)wi;
        const float4* b  = (const float4*)wj;
        float4* sa = (float4*)&sW[0][0];
        float4* sb = (float4*)&sW[1][0];
        for (int t = threadIdx.x; t < (N_PROP * N_PROP) / 4; t += BLOCK_THREADS) {
            sa[t] = a[t];
            sb[t] = b[t];
        }
    }
    __syncthreads();

    const int rows   = nPairs * 3;
    const int nTiles = (rows + 15) >> 4;

    const int lane  = threadIdx.x & 31;
    const int laneM = lane & 15;          // M (A/C) or N (B) index within tile
    const int koff  = (lane >> 4) << 1;   // K offset within 4-wide k-step: 0 or 2
    const int mhi   = (lane >> 4) << 3;   // C/D row offset: lanes 16-31 hold M+8

    const int waveId = blockIdx.x * WAVES_PER_BLOCK + (threadIdx.x >> 5);
    const int nWaves = gridDim.x * WAVES_PER_BLOCK;

    for (int tile = waveId; tile < nTiles; tile += nWaves) {
        const int r  = tile * 16 + laneM;
        const int rc = (r < rows) ? r : (rows - 1);   // clamp (rows % 16 == 0 in practice)
        const int p  = rc / 3;
        const int x  = rc - p * 3;
        const int ia = __builtin_nontemporal_load(&ind2[2 * p + 0]); // read-once stream
        const int ja = __builtin_nontemporal_load(&ind2[2 * p + 1]);
        const float* rowI = px + ((size_t)ia * 3 + x) * N_PROP;
        const float* rowJ = px + ((size_t)ja * 3 + x) * N_PROP;

        // ---- Batch-prefetch ALL A fragments for this tile (32 x b64 loads) ----
        v2f aI[16], aJ[16];
        #pragma unroll
        for (int kk = 0; kk < 16; ++kk) {
            const int z0 = (kk << 2) + koff;          // even -> 8B-aligned pair
            aI[kk] = *(const v2f*)(rowI + z0);
            aJ[kk] = *(const v2f*)(rowJ + z0);
        }

        v8f acc[4];
        #pragma unroll
        for (int n = 0; n < 4; ++n) acc[n] = v8f{0.f,0.f,0.f,0.f,0.f,0.f,0.f,0.f};

        // ---- 128 WMMAs; B fragments are loop-invariant (LDS -> hoisted regs) ----
        #pragma unroll
        for (int s = 0; s < 2; ++s) {
            const float* w = &sW[s][0];
            #pragma unroll
            for (int kk = 0; kk < 16; ++kk) {
                const int z0 = (kk << 2) + koff;
                const v2f a = s ? aJ[kk] : aI[kk];
                #pragma unroll
                for (int n = 0; n < 4; ++n) {
                    v2f b;
                    b.x = w[(z0    ) * N_PROP + n * 16 + laneM];
                    b.y = w[(z0 + 1) * N_PROP + n * 16 + laneM];
                    acc[n] = __builtin_amdgcn_wmma_f32_16x16x4_f32(
                        /*neg_a=*/false, a, /*neg_b=*/false, b,
                        /*c_mod=*/(short)0, acc[n],
                        /*reuse_a=*/false, /*reuse_b=*/false);
                }
            }
        }

        // ---- Store 16x64 tile: NT hint keeps the 614 MB stream out of L2 ----
        // C/D layout: VGPR v -> row (tile*16 + mhi + v), laneM -> column.
        float* ob = out + (size_t)tile * 16 * N_PROP;
        if (tile * 16 + 15 < rows) {
            #pragma unroll
            for (int n = 0; n < 4; ++n) {
                const int col = n * 16 + laneM;
                #pragma unroll
                for (int v = 0; v < 8; ++v)
                    __builtin_nontemporal_store(acc[n][v],
                        &ob[(size_t)(mhi + v) * N_PROP + col]);
            }
        } else {
            #pragma unroll
            for (int n = 0; n < 4; ++n) {
                const int col = n * 16 + laneM;
                #pragma unroll
                for (int v = 0; v < 8; ++v) {
                    const int rr = tile * 16 + mhi + v;
                    if (rr < rows)
                        __builtin_nontemporal_store(acc[n][v],
                            &out[(size_t)rr * N_PROP + col]);
                }
            }
        }
    }
}

extern "C" void kernel_launch(void* const* d_in, const int* in_sizes, int n_in,
                              void* d_out, int out_size, void* d_ws, size_t ws_size,
                              hipStream_t stream) {
    const int*   ind2 = (const int*)d_in[0];   // int32 pair indices
    const float* px   = (const float*)d_in[1]; // fp32 [nAtoms,3,64]
    const float* wi   = (const float*)d_in[2]; // fp32 [64,64]
    const float* wj   = (const float*)d_in[3]; // fp32 [64,64]
    float*       out  = (float*)d_out;         // fp32 [nPairs*3,64]

    const int nPairs = in_sizes[0] / 2;
    const int nTiles = (nPairs * 3 + 15) / 16;
    int blocks = (nTiles + WAVES_PER_BLOCK - 1) / WAVES_PER_BLOCK;
    if (blocks > 2048) blocks = 2048;
    if (blocks < 1) blocks = 1;

    pix_wmma_kernel<<<blocks, BLOCK_THREADS, 0, stream>>>(ind2, px, wi, wj, out, nPairs);
}